// Net_63866163691603
// MI455X (gfx1250) — compile-verified
//
#include <hip/hip_runtime.h>

// A3TGCN with H0==0: GRU collapses to H = (1-sigmoid(s*az+cz)) * tanh(s*ah+ch)
// where s[n,p] is the scalar symmetric-normalized GCN aggregation of x[:,0,p].
// Pipeline:
//   k_consts   : fold 1x4 conv + 8x4 lin weights into 16 scalars + softmax(att)
//   k_deg_init : deg = 1 (self loop)
//   k_deg_count: deg[dst] += 1 over edges (f32 atomics)
//   k_selfloop : dinv = rsqrt(deg); s[n,p] = dinv^2 * x[n,p]
//   k_edge_agg : s[dst,p] += dinv[src]*dinv[dst]*x[src,p] (12 f32 atomics/edge)
//   k_gru      : hacc[n,c] = sum_p probs[p]*(1-z)*h
//   k_out_wmma : out[n,0:12] = hacc[n,0:4] @ out_w + out_b via V_WMMA_F32_16X16X4_F32

typedef float v2f __attribute__((ext_vector_type(2)));
typedef float v8f __attribute__((ext_vector_type(8)));

__global__ void k_consts(const float* __restrict__ czw, const float* __restrict__ czb,
                         const float* __restrict__ lzw, const float* __restrict__ lzb,
                         const float* __restrict__ chw, const float* __restrict__ chb,
                         const float* __restrict__ lhw, const float* __restrict__ lhb,
                         const float* __restrict__ att, float* __restrict__ cst) {
  if (threadIdx.x == 0 && blockIdx.x == 0) {
    for (int c = 0; c < 4; ++c) {
      float az = 0.f, cz = lzb[c], ah = 0.f, ch = lhb[c];
      for (int k = 0; k < 4; ++k) {
        az += czw[k] * lzw[k * 4 + c];   // lin_z_w rows 0..3 (H0 part is zero)
        cz += czb[k] * lzw[k * 4 + c];
        ah += chw[k] * lhw[k * 4 + c];
        ch += chb[k] * lhw[k * 4 + c];
      }
      cst[0 + c] = az; cst[4 + c] = cz; cst[8 + c] = ah; cst[12 + c] = ch;
    }
    float mx = att[0];
    for (int p = 1; p < 12; ++p) mx = fmaxf(mx, att[p]);
    float e[12], sum = 0.f;
    for (int p = 0; p < 12; ++p) { e[p] = __expf(att[p] - mx); sum += e[p]; }
    float inv = 1.f / sum;
    for (int p = 0; p < 12; ++p) cst[16 + p] = e[p] * inv;
  }
}

__global__ void k_deg_init(float* __restrict__ deg, int N) {
  int n = blockIdx.x * blockDim.x + threadIdx.x;
  if (n < N) deg[n] = 1.0f;  // self loop
}

__global__ void k_deg_count(const int* __restrict__ dst, float* __restrict__ deg, int E) {
  int e = blockIdx.x * blockDim.x + threadIdx.x;
  if (e < E) atomicAdd(&deg[dst[e]], 1.0f);
}

__global__ void k_selfloop(const float* __restrict__ x, float* __restrict__ dinv,
                           float* __restrict__ s, int N) {
  int n = blockIdx.x * blockDim.x + threadIdx.x;
  if (n >= N) return;
  float d = dinv[n];              // currently holds degree (>=1)
  float di = rsqrtf(d);
  dinv[n] = di;
  float w = di * di;
  const float4* xr = (const float4*)(x + (size_t)n * 12);
  float4 a = xr[0], b = xr[1], c = xr[2];
  float4* so = (float4*)(s + (size_t)n * 12);
  so[0] = make_float4(w * a.x, w * a.y, w * a.z, w * a.w);
  so[1] = make_float4(w * b.x, w * b.y, w * b.z, w * b.w);
  so[2] = make_float4(w * c.x, w * c.y, w * c.z, w * c.w);
}

__global__ void k_edge_agg(const int* __restrict__ src, const int* __restrict__ dst,
                           const float* __restrict__ dinv, const float* __restrict__ x,
                           float* __restrict__ s, int E) {
  int e = blockIdx.x * blockDim.x + threadIdx.x;
  if (e >= E) return;
  int u = src[e], v = dst[e];
  float w = dinv[u] * dinv[v];
  const float4* xr = (const float4*)(x + (size_t)u * 12);
  float4 a = xr[0], b = xr[1], c = xr[2];
  float* so = s + (size_t)v * 12;
  atomicAdd(so + 0,  w * a.x); atomicAdd(so + 1,  w * a.y);
  atomicAdd(so + 2,  w * a.z); atomicAdd(so + 3,  w * a.w);
  atomicAdd(so + 4,  w * b.x); atomicAdd(so + 5,  w * b.y);
  atomicAdd(so + 6,  w * b.z); atomicAdd(so + 7,  w * b.w);
  atomicAdd(so + 8,  w * c.x); atomicAdd(so + 9,  w * c.y);
  atomicAdd(so + 10, w * c.z); atomicAdd(so + 11, w * c.w);
}

__global__ void k_gru(const float* __restrict__ s, const float* __restrict__ cst,
                      float* __restrict__ hacc, int N) {
  int n = blockIdx.x * blockDim.x + threadIdx.x;
  if (n >= N) return;
  float az[4], cz[4], ah[4], ch[4], pb[12];
#pragma unroll
  for (int c = 0; c < 4; ++c) {
    az[c] = cst[c]; cz[c] = cst[4 + c]; ah[c] = cst[8 + c]; ch[c] = cst[12 + c];
  }
#pragma unroll
  for (int p = 0; p < 12; ++p) pb[p] = cst[16 + p];
  const float4* sp4 = (const float4*)(s + (size_t)n * 12);
  float4 t0 = sp4[0], t1 = sp4[1], t2 = sp4[2];
  float sv[12] = {t0.x, t0.y, t0.z, t0.w, t1.x, t1.y, t1.z, t1.w, t2.x, t2.y, t2.z, t2.w};
  float acc0 = 0.f, acc1 = 0.f, acc2 = 0.f, acc3 = 0.f;
#pragma unroll
  for (int p = 0; p < 12; ++p) {
    float sp_ = sv[p], w = pb[p];
    float a[4];
#pragma unroll
    for (int c = 0; c < 4; ++c) {
      float tz = fminf(fmaxf(sp_ * az[c] + cz[c], -20.f), 20.f);
      float z = 1.f / (1.f + __expf(-tz));
      float th = fminf(fmaxf(sp_ * ah[c] + ch[c], -10.f), 10.f);
      float e2 = __expf(2.f * th);
      float h = (e2 - 1.f) / (e2 + 1.f);
      a[c] = w * (1.f - z) * h;
    }
    acc0 += a[0]; acc1 += a[1]; acc2 += a[2]; acc3 += a[3];
  }
  float4* hp = (float4*)(hacc + (size_t)n * 4);
  *hp = make_float4(acc0, acc1, acc2, acc3);
}

// Final projection: out[n,0:12] = hacc[n,0:4] @ out_w[4,12] + out_b.
// One V_WMMA_F32_16X16X4_F32 per wave handles 16 nodes (M=16, K=4, N=12<=16).
__global__ __launch_bounds__(256) void k_out_wmma(
    const float* __restrict__ hacc, const float* __restrict__ ow,
    const float* __restrict__ ob, float* __restrict__ out, int N) {
  int lane = threadIdx.x & 31;
  int wave = blockIdx.x * (blockDim.x >> 5) + (threadIdx.x >> 5);
  int nodeBase = wave * 16;
  if (nodeBase >= N) return;  // uniform per wave: EXEC stays all-ones for WMMA

  int m = lane & 15;          // A row / D column within half
  int khalf = lane >> 4;      // 0 -> K=0,1 ; 1 -> K=2,3 (ISA 32-bit A 16x4 layout)

  // A: 16x4 tile of hacc, 2 VGPRs per lane
  int node = nodeBase + m;
  int nodeC = node < N ? node : (N - 1);
  const float2* ap = (const float2*)(hacc + (size_t)nodeC * 4 + khalf * 2);
  float2 av = *ap;
  v2f a; a[0] = av.x; a[1] = av.y;

  // B: 4x16 (K x N), N=col=m, rows khalf*2 and khalf*2+1; pad cols 12..15 with 0
  v2f b; b[0] = 0.f; b[1] = 0.f;
  float bias = 0.f;
  if (m < 12) {
    b[0] = ow[(khalf * 2 + 0) * 12 + m];
    b[1] = ow[(khalf * 2 + 1) * 12 + m];
    bias = ob[m];
  }

  // C: bias broadcast down each column
  v8f c;
#pragma unroll
  for (int r = 0; r < 8; ++r) c[r] = bias;

  v8f d = __builtin_amdgcn_wmma_f32_16x16x4_f32(
      /*neg_a=*/false, a, /*neg_b=*/false, b,
      /*c_mod=*/(short)0, c, /*reuse_a=*/false, /*reuse_b=*/false);

  // D: VGPR r -> row nodeBase + r + khalf*8, col m. Wave-uniform full-tile fast
  // path (plain stores, immediate offsets); per-row guard only in the tail wave.
  float* op = out + (size_t)(nodeBase + khalf * 8) * 12 + m;
  if (nodeBase + 16 <= N) {         // uniform branch
    if (m < 12) {
#pragma unroll
      for (int r = 0; r < 8; ++r) op[r * 12] = d[r];
    }
  } else {
    if (m < 12) {
#pragma unroll
      for (int r = 0; r < 8; ++r) {
        if (nodeBase + khalf * 8 + r < N) op[r * 12] = d[r];
      }
    }
  }
}

extern "C" void kernel_launch(void* const* d_in, const int* in_sizes, int n_in,
                              void* d_out, int out_size, void* d_ws, size_t ws_size,
                              hipStream_t stream) {
  const float* x    = (const float*)d_in[0];   // [N,1,12]
  const int*   ei   = (const int*)d_in[1];     // [2,E]
  const float* czw  = (const float*)d_in[2];
  const float* czb  = (const float*)d_in[3];
  const float* lzw  = (const float*)d_in[4];
  const float* lzb  = (const float*)d_in[5];
  // d_in[6..9] (conv_r / lin_r) are dead: r only multiplies H0 == 0
  const float* chw  = (const float*)d_in[10];
  const float* chb  = (const float*)d_in[11];
  const float* lhw  = (const float*)d_in[12];
  const float* lhb  = (const float*)d_in[13];
  const float* att  = (const float*)d_in[14];
  const float* ow   = (const float*)d_in[15];  // [4,12]
  const float* ob   = (const float*)d_in[16];  // [12]
  float* out = (float*)d_out;

  int N = in_sizes[0] / 12;
  int E = in_sizes[1] / 2;
  const int* src = ei;
  const int* dst = ei + E;

  // workspace layout
  float* cst  = (float*)d_ws;                 // 64 floats of folded constants
  float* dinv = cst + 64;                     // [N] degree -> rsqrt(degree)
  float* s    = dinv + N;                     // [N,12] aggregated scalar per period
  float* hacc = s + (size_t)N * 12;           // [N,4] attention-weighted hidden

  const int BT = 256;
  int nBlkN = (N + BT - 1) / BT;
  int nBlkE = (E + BT - 1) / BT;

  k_consts<<<1, 32, 0, stream>>>(czw, czb, lzw, lzb, chw, chb, lhw, lhb, att, cst);
  k_deg_init<<<nBlkN, BT, 0, stream>>>(dinv, N);
  k_deg_count<<<nBlkE, BT, 0, stream>>>(dst, dinv, E);
  k_selfloop<<<nBlkN, BT, 0, stream>>>(x, dinv, s, N);
  k_edge_agg<<<nBlkE, BT, 0, stream>>>(src, dst, dinv, x, s, E);
  k_gru<<<nBlkN, BT, 0, stream>>>(s, cst, hacc, N);

  int waves = (N + 15) / 16;                  // one 16-node WMMA tile per wave
  int wBlk = (waves + 7) / 8;                 // 256 threads = 8 wave32 per block
  k_out_wmma<<<wBlk, BT, 0, stream>>>(hacc, ow, ob, out, N);
}